// StieltjesTransform_24842090840437
// MI455X (gfx1250) — compile-verified
//
#include <hip/hip_runtime.h>
#include <hip/hip_bf16.h>

#define ROW_N   2048        // reduction length (dim=3 of [2,16,2048,2048])
#define WPB     4           // waves per block (128 threads)
#define CHUNKS  16          // float4 chunks per lane: 16 * 32 lanes * 4 floats = 2048
#define EPSF    1e-9f

// ---- helpers -------------------------------------------------------------

__device__ __forceinline__ float clamp50(float v) {
    return fminf(fmaxf(v, -50.0f), 50.0f);
}

// fast reciprocal for the bisection sum (v_rcp_f32, ~1 ulp)
__device__ __forceinline__ float fast_rcp(float d) {
    return __builtin_amdgcn_rcpf(d);
}

// accurate reciprocal for the final output: rcp + one Newton step
__device__ __forceinline__ float acc_rcp(float d) {
    float r = __builtin_amdgcn_rcpf(d);
    return fmaf(r, fmaf(-d, r, 1.0f), r);
}

// ---- kernel --------------------------------------------------------------

__global__ __launch_bounds__(WPB * 32)
void stieltjes_kernel(const float* __restrict__ x, float* __restrict__ y,
                      const int* __restrict__ qptr, const int* __restrict__ iptr,
                      int nrows)
{
    // per-wave private double buffer: 4 waves * 2 * 8KB = 64KB LDS / block
    __shared__ float4 buf[WPB][2][ROW_N / 4];

    const int lane  = threadIdx.x & 31;
    const int wid   = threadIdx.x >> 5;
    const int gwave = blockIdx.x * WPB + wid;
    const int nwav  = gridDim.x * WPB;

    const int   q       = qptr[0];   // uniform scalar load
    const int   niter   = iptr[0];
    const float nq      = -(float)q;
    const float ub_init = (q == 1) ? (float)ROW_N
                                   : powf((float)ROW_N, 1.0f / (float)q);

    // CDNA5 async copy: stage one row (8KB) into this wave's LDS slice.
    // 16 x global_load_async_to_lds_b128 => 16 ASYNCcnt increments, in-order.
    auto prefetch = [&](int row, int p) {
        const float* rp = x + (size_t)row * ROW_N;
        unsigned lbase = (unsigned)(unsigned long long)(&buf[wid][p][0]);
#pragma unroll
        for (int c = 0; c < CHUNKS; ++c) {
            const int    f = c * 32 + lane;         // float4 index in row
            const float* g = rp + f * 4;
            unsigned     l = lbase + f * 16;
            asm volatile("global_load_async_to_lds_b128 %0, %1, off"
                         :: "v"(l), "v"(g) : "memory");
        }
    };

    int p = 0;
    if (gwave < nrows) prefetch(gwave, 0);

    for (int row = gwave; row < nrows; row += nwav) {
        const int nxt = row + nwav;
        if (nxt < nrows) {
            prefetch(nxt, p ^ 1);                       // overlap with compute
            asm volatile("s_wait_asynccnt 0x10" ::: "memory");  // cur row done
        } else {
            asm volatile("s_wait_asynccnt 0x0" ::: "memory");
        }

        // ---- load row from LDS into registers, clamp, lane-local max ----
        float4 t[CHUNKS];
        float  m = -50.0f;
#pragma unroll
        for (int c = 0; c < CHUNKS; ++c) {
            float4 v = buf[wid][p][c * 32 + lane];      // ds_load_b128
            v.x = clamp50(v.x); v.y = clamp50(v.y);
            v.z = clamp50(v.z); v.w = clamp50(v.w);
            t[c] = v;
            m = fmaxf(m, fmaxf(fmaxf(v.x, v.y), fmaxf(v.z, v.w)));
        }
        // wave32 butterfly max: all lanes end with the row max
#pragma unroll
        for (int s = 16; s > 0; s >>= 1)
            m = fmaxf(m, __shfl_xor(m, s, 32));

        // t_j = x_max - x_j >= 0 exactly (exact negation of reference's x_i,
        // so mid + t is bit-identical to the reference's mid - x_i)
#pragma unroll
        for (int c = 0; c < CHUNKS; ++c) {
            t[c].x = m - t[c].x; t[c].y = m - t[c].y;
            t[c].z = m - t[c].z; t[c].w = m - t[c].w;
        }

        // ---- bisection for lambda ----
        float lb = EPSF, ub = ub_init;
        for (int it = 0; it < niter; ++it) {
            const float mid = 0.5f * (lb + ub);
            float a0 = 0.f, a1 = 0.f, a2 = 0.f, a3 = 0.f;  // 4 accumulators (ILP)
            if (q == 1) {
                // HOT PATH. Note: no eps-clamp needed -- mid >= lb >= EPS and
                // t >= 0, and fp add of nonnegatives never rounds below either
                // operand, so max(mid + t, EPS) == mid + t bit-exactly.
#pragma unroll
                for (int c = 0; c < CHUNKS; ++c) {
                    a0 += fast_rcp(mid + t[c].x);
                    a1 += fast_rcp(mid + t[c].y);
                    a2 += fast_rcp(mid + t[c].z);
                    a3 += fast_rcp(mid + t[c].w);
                }
            } else {
                // COLD general-q path: compact (not unrolled), re-read from LDS
                // so the hot path's register file & I-cache stay small.
#pragma unroll 1
                for (int c = 0; c < CHUNKS; ++c) {
                    float4 v = buf[wid][p][c * 32 + lane];
                    a0 += powf(fmaxf(mid + (m - clamp50(v.x)), EPSF), nq);
                    a1 += powf(fmaxf(mid + (m - clamp50(v.y)), EPSF), nq);
                    a2 += powf(fmaxf(mid + (m - clamp50(v.z)), EPSF), nq);
                    a3 += powf(fmaxf(mid + (m - clamp50(v.w)), EPSF), nq);
                }
            }
            float s = (a0 + a1) + (a2 + a3);
            // butterfly sum: identical total in every lane -> uniform lb/ub
#pragma unroll
            for (int o = 16; o > 0; o >>= 1)
                s += __shfl_xor(s, o, 32);
            const bool gt = (s - 1.0f) > 0.0f;
            lb = gt ? mid : lb;
            ub = gt ? ub : mid;
        }
        const float lam = 0.5f * (lb + ub);

        // ---- output: max(lam - x_i, eps)^(-q), coalesced b128 stores ----
        float* op = y + (size_t)row * ROW_N;
        if (q == 1) {
#pragma unroll
            for (int c = 0; c < CHUNKS; ++c) {
                float4 r;
                r.x = acc_rcp(fmaxf(lam + t[c].x, EPSF));
                r.y = acc_rcp(fmaxf(lam + t[c].y, EPSF));
                r.z = acc_rcp(fmaxf(lam + t[c].z, EPSF));
                r.w = acc_rcp(fmaxf(lam + t[c].w, EPSF));
                *reinterpret_cast<float4*>(op + (c * 32 + lane) * 4) = r;
            }
        } else {
#pragma unroll 1
            for (int c = 0; c < CHUNKS; ++c) {
                float4 v = buf[wid][p][c * 32 + lane];
                float4 r;
                r.x = powf(fmaxf(lam + (m - clamp50(v.x)), EPSF), nq);
                r.y = powf(fmaxf(lam + (m - clamp50(v.y)), EPSF), nq);
                r.z = powf(fmaxf(lam + (m - clamp50(v.z)), EPSF), nq);
                r.w = powf(fmaxf(lam + (m - clamp50(v.w)), EPSF), nq);
                *reinterpret_cast<float4*>(op + (c * 32 + lane) * 4) = r;
            }
        }
        p ^= 1;
    }
}

// ---- launch --------------------------------------------------------------

extern "C" void kernel_launch(void* const* d_in, const int* in_sizes, int n_in,
                              void* d_out, int out_size, void* d_ws, size_t ws_size,
                              hipStream_t stream) {
    const float* logits = (const float*)d_in[0];
    // d_in[1] = dim (always the last axis per reference) -- unused
    const int* qptr = (const int*)d_in[2];
    const int* iptr = (const int*)d_in[3];
    float*     out  = (float*)d_out;

    const int nrows = in_sizes[0] / ROW_N;     // 65536 rows for the reference shape

    int blocks = (nrows + WPB - 1) / WPB;
    if (blocks > 2048) blocks = 2048;          // persistent waves: 8192 waves, ~8 rows each
    if (blocks < 1) blocks = 1;

    stieltjes_kernel<<<blocks, WPB * 32, 0, stream>>>(logits, out, qptr, iptr, nrows);
}